// DGLGATNE_5634997092481
// MI455X (gfx1250) — compile-verified
//
#include <hip/hip_runtime.h>
#include <hip/hip_bf16.h>
#include <math.h>

// Problem constants (match reference)
#define EMB     200
#define U_DIM   10
#define T_DIM   4
#define DIM_A   20
#define N_SRC   65536
#define N_DST   8192
#define E_PER_T 262144     // 2^18
#define KPAD    12         // K=10 padded to 12 -> 3x K=4 f32 WMMA steps

typedef __attribute__((ext_vector_type(2))) float v2f;
typedef __attribute__((ext_vector_type(8))) float v8f;

// ---------------- kernel 0: zero the scatter accumulator ----------------
__global__ void k_zero(float* __restrict__ p, int n) {
  int i = blockIdx.x * blockDim.x + threadIdx.x;
  if (i < n) p[i] = 0.0f;
}

// ---------------- kernel 1: edge scatter-add ----------------------------
// agg[t][d][u] += node_type_embeddings[input_nodes[edge_src[t,e]]][t][u]
__global__ void k_scatter(const int* __restrict__ edge_src,
                          const int* __restrict__ edge_dst,
                          const int* __restrict__ input_nodes,
                          const float* __restrict__ nte_tbl,   // [NUM_NODES, T, U]
                          float* __restrict__ agg)             // [T, N_DST, U]
{
  int idx = blockIdx.x * blockDim.x + threadIdx.x;  // 0 .. T*E-1 (exact grid)
  int t = idx >> 18;                 // E_PER_T == 2^18
  int e = idx & (E_PER_T - 1);
  int s = edge_src[t * E_PER_T + e];
  int d = edge_dst[t * E_PER_T + e];
  int node = input_nodes[s];
  const float* src = nte_tbl + (size_t)node * (T_DIM * U_DIM) + t * U_DIM;
  float* dst = agg + ((size_t)t * N_DST + d) * U_DIM;
#pragma unroll
  for (int u = 0; u < U_DIM; ++u)
    atomicAdd(dst + u, src[u]);      // global_atomic_add_f32, L2-resident region
}

// ---------------- kernel 2: per-dst attention ---------------------------
// h = tanh(nte @ s1), score = h @ s2, att = softmax_t, combined = att @ nte.
// combined stored padded to KPAD with explicit zeros (exact WMMA K-tail).
__global__ void k_attn(const float* __restrict__ agg,  // [T, N_DST, U]
                       const float* __restrict__ s1,   // [T, U, DIM_A]
                       const float* __restrict__ s2,   // [T, DIM_A]
                       float* __restrict__ cbuf)       // [N_DST, KPAD]
{
  __shared__ float ls1[T_DIM * U_DIM * DIM_A];  // 800 floats
  __shared__ float ls2[T_DIM * DIM_A];          // 80 floats
  int tid = threadIdx.x;
  for (int i = tid; i < T_DIM * U_DIM * DIM_A; i += blockDim.x) ls1[i] = s1[i];
  for (int i = tid; i < T_DIM * DIM_A; i += blockDim.x)         ls2[i] = s2[i];
  __syncthreads();

  int b = blockIdx.x * blockDim.x + tid;        // grid is exactly N_DST threads

  float nte[T_DIM][U_DIM];
#pragma unroll
  for (int t = 0; t < T_DIM; ++t) {
    const float* p = agg + ((size_t)t * N_DST + b) * U_DIM;
#pragma unroll
    for (int u = 0; u < U_DIM; ++u) nte[t][u] = p[u];
  }

  float score[T_DIM];
#pragma unroll
  for (int t = 0; t < T_DIM; ++t) {
    float sc = 0.0f;
    for (int a = 0; a < DIM_A; ++a) {
      float h = 0.0f;
#pragma unroll
      for (int u = 0; u < U_DIM; ++u)
        h = fmaf(nte[t][u], ls1[t * U_DIM * DIM_A + u * DIM_A + a], h);
      sc = fmaf(tanhf(h), ls2[t * DIM_A + a], sc);
    }
    score[t] = sc;
  }

  float m = score[0];
#pragma unroll
  for (int t = 1; t < T_DIM; ++t) m = fmaxf(m, score[t]);
  float ex[T_DIM];
  float den = 0.0f;
#pragma unroll
  for (int t = 0; t < T_DIM; ++t) { ex[t] = expf(score[t] - m); den += ex[t]; }
  float inv = 1.0f / den;

  float* cp = cbuf + (size_t)b * KPAD;
#pragma unroll
  for (int u = 0; u < U_DIM; ++u) {
    float c = 0.0f;
#pragma unroll
    for (int t = 0; t < T_DIM; ++t) c = fmaf(ex[t] * inv, nte[t][u], c);
    cp[u] = c;
  }
  cp[10] = 0.0f;   // K padding -> WMMA k=10,11 terms are exactly zero
  cp[11] = 0.0f;
}

// ---------------- kernel 3: WMMA f32 projection + bias ------------------
// out_pre[b, t, e] = node_emb[output_nodes[b], e] + sum_u combined[b,u]*W[t,u,e]
// as a [8192 x 12] x [12 x 800] GEMM using V_WMMA_F32_16X16X4_F32 (3 K-steps).
// NOTE: no guards/branches -> EXEC stays all-1s as WMMA requires.
__global__ void k_wmma_proj(const float* __restrict__ cbuf,        // [N_DST, KPAD]
                            const float* __restrict__ W,           // [T, U, EMB]
                            const int*   __restrict__ output_nodes,
                            const float* __restrict__ node_emb,    // [NUM_NODES, EMB]
                            float* __restrict__ out)               // [N_DST, T, EMB]
{
  const int lane = threadIdx.x;       // 0..31 (wave32)
  const int half = lane >> 4;         // lane group 0-15 / 16-31
  const int lm   = lane & 15;
  const int m0 = (blockIdx.y * blockDim.y + threadIdx.y) * 16;  // 512 M-tiles
  const int nb = blockIdx.x * 16;                               // 50  N-tiles

  // A (16x4 f32 per step): VGPR0 = A[m, half*2], VGPR1 = A[m, half*2+1]
  const float* arow = cbuf + (size_t)(m0 + lm) * KPAD + half * 2;
  v2f a0, a1, a2;
  a0.x = arow[0]; a0.y = arow[1];
  a1.x = arow[4]; a1.y = arow[5];
  a2.x = arow[8]; a2.y = arow[9];

  // B (4x16 f32 per step): VGPR0 = B[half*2, n], VGPR1 = B[half*2+1, n]
  // B[k, n] = W[t, u=k, e] with n = t*EMB + e
  const int n = nb + lm;              // < 800 always
  const int t = n / EMB;
  const int e = n - t * EMB;
  const float* wcol = W + (size_t)t * (U_DIM * EMB) + e;  // + k*EMB per row
  const int kb = half * 2;
  v2f b0, b1, b2;
  b0.x = wcol[(kb + 0) * EMB]; b0.y = wcol[(kb + 1) * EMB];
  b1.x = wcol[(kb + 4) * EMB]; b1.y = wcol[(kb + 5) * EMB];
  // k = 8+kb+{0,1}: for half==1 that is k=10,11 (out of the U=10 range).
  // Select a safe in-range address and mask the value to 0 (v_cndmask, no
  // EXEC divergence). The matching A values are zero anyway (KPAD zeros).
  {
    const float msk = (half == 0) ? 1.0f : 0.0f;
    const int k8 = (half == 0) ? 8 : 0;
    const int k9 = (half == 0) ? 9 : 1;
    b2.x = wcol[k8 * EMB] * msk;
    b2.y = wcol[k9 * EMB] * msk;
  }

  v8f c = {};
  c = __builtin_amdgcn_wmma_f32_16x16x4_f32(false, a0, false, b0, (short)0, c, false, false);
  c = __builtin_amdgcn_wmma_f32_16x16x4_f32(false, a1, false, b1, (short)0, c, false, false);
  c = __builtin_amdgcn_wmma_f32_16x16x4_f32(false, a2, false, b2, (short)0, c, false, false);

  // D layout: VGPR i -> lanes 0-15: M=m0+i, lanes 16-31: M=m0+8+i; N = nb+lm.
  // Fuse the base-embedding gather into the store.
#pragma unroll
  for (int i = 0; i < 8; ++i) {
    const int b = m0 + half * 8 + i;
    const int on = output_nodes[b];
    const float base = node_emb[(size_t)on * EMB + e];
    out[((size_t)b * T_DIM + t) * EMB + e] = c[i] + base;
  }
}

// ---------------- kernel 4: L2-normalize each (b,t) row in place --------
__global__ void k_norm(float* __restrict__ out)   // [N_DST*T, EMB] rows
{
  const int lane = threadIdx.x;                                  // 32 lanes
  const int row  = blockIdx.x * blockDim.y + threadIdx.y;        // 32768 rows
  float* p = out + (size_t)row * EMB;

  float s = 0.0f;
  for (int i = lane; i < EMB; i += 32) { float v = p[i]; s = fmaf(v, v, s); }
#pragma unroll
  for (int off = 16; off > 0; off >>= 1)
    s += __shfl_xor(s, off, 32);                  // wave32 butterfly

  const float inv = 1.0f / fmaxf(sqrtf(s), 1e-12f);
  for (int i = lane; i < EMB; i += 32) p[i] *= inv;
}

extern "C" void kernel_launch(void* const* d_in, const int* in_sizes, int n_in,
                              void* d_out, int out_size, void* d_ws, size_t ws_size,
                              hipStream_t stream) {
  (void)in_sizes; (void)n_in; (void)out_size; (void)ws_size;

  const int*   input_nodes  = (const int*)  d_in[0];   // [N_SRC]
  const int*   output_nodes = (const int*)  d_in[1];   // [N_DST]
  const int*   edge_src     = (const int*)  d_in[2];   // [T, E]
  const int*   edge_dst     = (const int*)  d_in[3];   // [T, E]
  const float* node_emb     = (const float*)d_in[4];   // [NUM_NODES, EMB]
  const float* nte_tbl      = (const float*)d_in[5];   // [NUM_NODES, T, U]
  const float* W            = (const float*)d_in[6];   // [T, U, EMB]
  const float* s1           = (const float*)d_in[7];   // [T, U, DIM_A]
  const float* s2           = (const float*)d_in[8];   // [T, DIM_A, 1]
  float* out = (float*)d_out;                          // [N_DST, T, EMB]

  // Workspace layout (~1.7 MB total)
  float* agg  = (float*)d_ws;                          // [T, N_DST, U]
  float* cbuf = agg + (size_t)T_DIM * N_DST * U_DIM;   // [N_DST, KPAD]

  const int agg_n = T_DIM * N_DST * U_DIM;             // 327680
  k_zero<<<(agg_n + 255) / 256, 256, 0, stream>>>(agg, agg_n);

  k_scatter<<<(T_DIM * E_PER_T) / 256, 256, 0, stream>>>(
      edge_src, edge_dst, input_nodes, nte_tbl, agg);

  k_attn<<<N_DST / 256, 256, 0, stream>>>(agg, s1, s2, cbuf);

  // 50 N-tiles x 512 M-tiles; 4 waves per block, one 16x16 tile per wave
  k_wmma_proj<<<dim3(50, 128), dim3(32, 4), 0, stream>>>(
      cbuf, W, output_nodes, node_emb, out);

  // one wave per (b,t) row, 8 waves per block
  k_norm<<<(N_DST * T_DIM) / 8, dim3(32, 8), 0, stream>>>(out);
}